// AdaptiveComputationTime_90194313216433
// MI455X (gfx1250) — compile-verified
//
#include <hip/hip_runtime.h>
#include <hip/hip_bf16.h>
#include <math.h>

// Problem constants (match reference)
constexpr int B_ = 16;
constexpr int M_ = 2048;
constexpr int H_ = 1024;
constexpr float THRESH = 0.99f;

typedef __attribute__((ext_vector_type(2))) float v2f;
typedef __attribute__((ext_vector_type(4))) float v4f;
typedef __attribute__((ext_vector_type(8))) float v8f;

// ---------------------------------------------------------------------------
// K1: per-batch-row scan of `run` -> gather (unpack) indices and live count.
// One block (256 threads) per row; each thread owns 8 consecutive tokens.
// ---------------------------------------------------------------------------
__global__ void k_scan_run(const int* __restrict__ run,
                           int* __restrict__ gatherIdx,
                           int* __restrict__ liveCnt0) {
    __shared__ int sScan[256];
    const int b = blockIdx.x;
    const int t = threadIdx.x;
    const int base = b * M_;

    int vals[8];
    int tot = 0;
#pragma unroll
    for (int j = 0; j < 8; ++j) {
        const int m = t * 8 + j;
        const int r = (run[base + m] != 0) ? 1 : 0;
        vals[j] = r;
        tot += r;
    }
    sScan[t] = tot;
    __syncthreads();
    for (int off = 1; off < 256; off <<= 1) {
        int x = (t >= off) ? sScan[t - off] : 0;
        __syncthreads();
        sScan[t] += x;
        __syncthreads();
    }
    int running = sScan[t] - tot;   // exclusive prefix of this thread
#pragma unroll
    for (int j = 0; j < 8; ++j) {
        const int m = t * 8 + j;
        gatherIdx[base + m] = vals[j] ? running : -1;
        running += vals[j];
    }
    if (t == 255) liveCnt0[b] = sScan[255];
}

// ---------------------------------------------------------------------------
// K2: logits via V_WMMA_F32_16X16X4_F32.
// Each wave owns a 16-token tile; loops K over H in steps of 4, weight vector
// placed in column 0 of B.  D column 0 (lanes 0 and 16) holds the 16 dots.
// Branch-free B construction: all lanes read LDS, masked by a 0/1 float, so
// the loop body has no EXEC divergence.  Two accumulators hide WMMA latency.
// Block = 256 threads = 8 waves = 128 tokens; grid = B*M/128 = 256 blocks.
// ---------------------------------------------------------------------------
__global__ void k_logits(const float* __restrict__ h,
                         const float* __restrict__ w,
                         const int* __restrict__ liveCnt0,
                         float* __restrict__ logits) {
    __shared__ float wlds[H_];
    const int t = threadIdx.x;
#pragma unroll
    for (int j = 0; j < H_ / 256; ++j) wlds[t + j * 256] = w[t + j * 256];
    __syncthreads();

    const int wave = t >> 5;
    const int lane = t & 31;
    const int tile = blockIdx.x * 8 + wave;   // 16-token tile id
    const int b    = tile >> 7;               // 128 tiles per row (M/16)
    const int i0   = (tile & 127) << 4;
    const int cnt0 = liveCnt0[b];
    if (i0 >= cnt0) return;                   // wave-uniform skip (EXEC stays full)

    const int  row  = i0 + (lane & 15);
    const int  koff = (lane >> 4) << 1;       // lanes 0-15: K=0,1 ; lanes 16-31: K=2,3
    const float* hrow = h + ((size_t)b * M_ + row) * H_ + koff;
    const bool l0 = (lane == 0), l16 = (lane == 16);
    const float act = (l0 || l16) ? 1.0f : 0.0f;  // only column 0 of B is live
    const int   o0  = l16 ? 1 : 0;                // B rows: lane0 gets k,k+2; lane16 k+1,k+3

    v8f acc0 = {0.f, 0.f, 0.f, 0.f, 0.f, 0.f, 0.f, 0.f};
    v8f acc1 = {0.f, 0.f, 0.f, 0.f, 0.f, 0.f, 0.f, 0.f};
    for (int k = 0; k < H_; k += 8) {
        v2f a0 = *(const v2f*)(hrow + k);          // global_load_b64
        v2f b0;
        b0.x = wlds[k + o0] * act;
        b0.y = wlds[k + o0 + 2] * act;
        acc0 = __builtin_amdgcn_wmma_f32_16x16x4_f32(
            false, a0, false, b0, (short)0, acc0, false, false);

        v2f a1 = *(const v2f*)(hrow + k + 4);
        v2f b1;
        b1.x = wlds[k + 4 + o0] * act;
        b1.y = wlds[k + 6 + o0] * act;
        acc1 = __builtin_amdgcn_wmma_f32_16x16x4_f32(
            false, a1, false, b1, (short)0, acc1, false, false);
    }
    v8f acc = acc0 + acc1;   // elementwise sum of partial D tiles

    // D column 0: lane 0 holds M=0..7 (vgpr r), lane 16 holds M=8..15
    if (l0) {
#pragma unroll
        for (int r = 0; r < 8; ++r) logits[b * M_ + i0 + r] = acc[r];
    } else if (l16) {
#pragma unroll
        for (int r = 0; r < 8; ++r) logits[b * M_ + i0 + 8 + r] = acc[r];
    }
}

// ---------------------------------------------------------------------------
// K3: per-row scalar logic + scan of run_new -> inverse pack index table.
// packIdx[b][j] = source row in h feeding packed slot j, or -1 (zero fill).
// ---------------------------------------------------------------------------
__global__ void k_scalars(const float* __restrict__ logits,
                          const float* __restrict__ acc_p,
                          const float* __restrict__ remainders,
                          const float* __restrict__ p_bias,
                          const int*   __restrict__ exit_in,
                          const int*   __restrict__ updates,
                          const int*   __restrict__ gatherIdx,
                          float* __restrict__ p_eff_ws,
                          int*   __restrict__ packIdx,
                          float* __restrict__ out_acc,
                          float* __restrict__ out_rem,
                          float* __restrict__ out_run,
                          float* __restrict__ out_exit) {
    __shared__ int sScan[256];
    __shared__ int sPack[M_];
    const int b = blockIdx.x;
    const int t = threadIdx.x;
    const int base = b * M_;
    const float bias = p_bias[0];
    const int upd1 = updates[0] + 1;

    int   rn[8];     // run_new bits
    int   srcv[8];   // unpack source index (valid when run_new)
    int   tot = 0;

#pragma unroll
    for (int j = 0; j < 8; ++j) {
        const int m   = t * 8 + j;
        const int idx = base + m;
        const int g   = gatherIdx[idx];
        const bool runb = (g >= 0);

        float p = 0.0f;
        if (runb) {
            const float lg = logits[base + g] + bias;
            p = 1.0f / (1.0f + expf(-lg));
        }
        const float accn = acc_p[idx] + p;
        const bool mc = (accn < THRESH) && runb;     // continue
        const bool me = (!mc) && runb;               // exit this step
        const float pe = mc ? p : (me ? (1.0f - (accn - p)) : 0.0f);

        out_acc[idx]  = accn;
        out_rem[idx]  = remainders[idx] + (me ? pe : 0.0f);
        out_run[idx]  = mc ? 1.0f : 0.0f;
        out_exit[idx] = (float)(exit_in[idx] + (me ? upd1 : 0));
        p_eff_ws[idx] = pe;

        rn[j]   = mc ? 1 : 0;
        srcv[j] = g;
        tot    += rn[j];
    }

    // block scan of run_new -> pack destinations
    sScan[t] = tot;
    __syncthreads();
    for (int off = 1; off < 256; off <<= 1) {
        int x = (t >= off) ? sScan[t - off] : 0;
        __syncthreads();
        sScan[t] += x;
        __syncthreads();
    }
    int dest = sScan[t] - tot;   // exclusive prefix

    // build inverse map in LDS: slot j -> h row index (or -1)
#pragma unroll
    for (int j = 0; j < 8; ++j) sPack[t * 8 + j] = -1;
    __syncthreads();
#pragma unroll
    for (int j = 0; j < 8; ++j) {
        if (rn[j]) sPack[dest] = srcv[j];
        dest += rn[j];
    }
    __syncthreads();
#pragma unroll
    for (int j = 0; j < 8; ++j) {
        const int m = t * 8 + j;
        packIdx[base + m] = sPack[m];
    }
}

// ---------------------------------------------------------------------------
// K4: bandwidth pass.  One block per (b, m) row; 256 threads x float4 = 1024.
//   weighted_h_new = hu * pe + wh * (1 - pe)   (hu gathered via gatherIdx)
//   h_packed[j]    = h[packIdx[j]] or 0
// wh and all outputs are streamed once -> non-temporal; h rows may be re-read
// (unpack + pack gathers) -> regular temporal so they can live in L2.
// ---------------------------------------------------------------------------
__global__ void k_big(const float* __restrict__ h,
                      const float* __restrict__ wh,
                      const float* __restrict__ p_eff_ws,
                      const int*   __restrict__ gatherIdx,
                      const int*   __restrict__ packIdx,
                      float* __restrict__ out_hp,
                      float* __restrict__ out_wh) {
    const int bm = blockIdx.x;            // 0 .. B*M-1
    const int b  = bm >> 11;              // / M_
    const int t  = threadIdx.x;
    const size_t rowOff = (size_t)bm * H_ + (size_t)t * 4;
    const size_t hbase  = (size_t)b * M_ * H_ + (size_t)t * 4;

    const float pe = p_eff_ws[bm];
    const int g  = gatherIdx[bm];
    const int pk = packIdx[bm];

    const v4f z = {0.f, 0.f, 0.f, 0.f};

    v4f wh4 = __builtin_nontemporal_load((const v4f*)(wh + rowOff));
    v4f hu4 = (g >= 0) ? *(const v4f*)(h + hbase + (size_t)g * H_) : z;
    const float om = 1.0f - pe;
    v4f o = hu4 * pe + wh4 * om;
    __builtin_nontemporal_store(o, (v4f*)(out_wh + rowOff));

    v4f hp4 = (pk >= 0) ? *(const v4f*)(h + hbase + (size_t)pk * H_) : z;
    __builtin_nontemporal_store(hp4, (v4f*)(out_hp + rowOff));
}

// ---------------------------------------------------------------------------
extern "C" void kernel_launch(void* const* d_in, const int* in_sizes, int n_in,
                              void* d_out, int out_size, void* d_ws, size_t ws_size,
                              hipStream_t stream) {
    const float* h      = (const float*)d_in[0];   // [B,M,H]
    const float* wh     = (const float*)d_in[1];   // [B,M,H]
    const float* acc_p  = (const float*)d_in[2];   // [B,M,1]
    const float* rem    = (const float*)d_in[3];   // [B,M,1]
    const float* p_w    = (const float*)d_in[4];   // [1,H]
    const float* p_b    = (const float*)d_in[5];   // [1]
    const int*   run    = (const int*)d_in[6];     // [B,M] bool
    const int*   exit_  = (const int*)d_in[7];     // [B,M,1] int32
    const int*   upd    = (const int*)d_in[8];     // scalar int

    const size_t BMH = (size_t)B_ * M_ * H_;
    const size_t BM  = (size_t)B_ * M_;

    float* out      = (float*)d_out;
    float* out_hp   = out;                 // h_packed
    float* out_wh   = out + BMH;           // weighted_h_new
    float* out_acc  = out + 2 * BMH;       // acc_new
    float* out_rem  = out_acc + BM;        // remainders_new
    float* out_run  = out_rem + BM;        // run_new (0/1 as float)
    float* out_exit = out_run + BM;        // exit_new (as float)

    float* logits   = (float*)d_ws;        // BM floats
    float* p_eff    = logits + BM;         // BM floats
    int*   gIdx     = (int*)(p_eff + BM);  // BM ints
    int*   pIdx     = gIdx + BM;           // BM ints
    int*   cnt0     = pIdx + BM;           // B ints

    k_scan_run<<<B_, 256, 0, stream>>>(run, gIdx, cnt0);

    k_logits<<<(B_ * (M_ / 16)) / 8, 256, 0, stream>>>(h, p_w, cnt0, logits);

    k_scalars<<<B_, 256, 0, stream>>>(logits, acc_p, rem, p_b, exit_, upd, gIdx,
                                      p_eff, pIdx, out_acc, out_rem, out_run, out_exit);

    k_big<<<B_ * M_, 256, 0, stream>>>(h, wh, p_eff, gIdx, pIdx, out_hp, out_wh);
}